// TensorProductScatter_80376017977771
// MI455X (gfx1250) — compile-verified
//
#include <hip/hip_runtime.h>
#include <stdint.h>

#define MULC   32
#define NFEAT  352
#define XROW   128   // 32 scalars + 32*3 vectors
#define WROW   160   // 5 * 32 weights
#define WAVES_PER_BLOCK 8
// per-wave staging buffer: 160 floats of weights + 128 floats of x row
#define STAGE_F (WROW + XROW)   // 288 floats = 1152 B, 16B-divisible

__device__ __forceinline__ float inv_sqrt3() { return 0.5773502691896258f; }
__device__ __forceinline__ float inv_sqrt2() { return 0.7071067811865476f; }

// ---------------- CSR build ----------------

__global__ void hist_kernel(const int* __restrict__ edge_dst,
                            int* __restrict__ hist, int E) {
    int i = blockIdx.x * blockDim.x + threadIdx.x;
    if (i < E) atomicAdd(&hist[edge_dst[i]], 1);
}

// Single-block Hillis-Steele scan over chunks of 1024 with running carry.
// Safe in-place: each chunk's reads complete before its writes.
__global__ void scan_kernel(int* __restrict__ hist_offs,   // in: hist, out: excl offsets (N+1)
                            int* __restrict__ cursor,      // out: copy of offsets (N)
                            int N) {
    __shared__ int buf[1024];
    __shared__ int carry_s;
    if (threadIdx.x == 0) carry_s = 0;
    __syncthreads();
    for (int base = 0; base < N; base += 1024) {
        int i = base + (int)threadIdx.x;
        int v = (i < N) ? hist_offs[i] : 0;
        buf[threadIdx.x] = v;
        __syncthreads();
        for (int off = 1; off < 1024; off <<= 1) {
            int t = (threadIdx.x >= (unsigned)off) ? buf[threadIdx.x - off] : 0;
            __syncthreads();
            buf[threadIdx.x] += t;
            __syncthreads();
        }
        int excl = buf[threadIdx.x] - v + carry_s;
        if (i < N) { hist_offs[i] = excl; cursor[i] = excl; }
        __syncthreads();
        if (threadIdx.x == 1023) carry_s += buf[1023];
        __syncthreads();
    }
    if (threadIdx.x == 0) hist_offs[N] = carry_s;
}

__global__ void scatter_kernel(const int* __restrict__ edge_dst,
                               int* __restrict__ cursor,
                               int* __restrict__ edge_ids, int E) {
    int i = blockIdx.x * blockDim.x + threadIdx.x;
    if (i < E) {
        int slot = atomicAdd(&cursor[edge_dst[i]], 1);
        edge_ids[slot] = i;
    }
}

// ---------------- async global -> LDS staging (gfx1250) ----------------
// Copies this edge's 640B weight row and 512B x[src] row into the wave's LDS
// stage buffer with three async transfers (ASYNCcnt-tracked, in-order done).
__device__ __forceinline__ void issue_stage(const float* __restrict__ ew_row,
                                            const float* __restrict__ x_row,
                                            uint32_t lds_base, int lane) {
    const char* wg  = (const char*)ew_row + lane * 16;        // bytes [0,512): w0..w3
    const char* w4g = (const char*)ew_row + 512 + lane * 4;   // bytes [512,640): w4
    const char* xg  = (const char*)x_row  + lane * 16;        // bytes [0,512): x row
    uint32_t lw  = lds_base + lane * 16;
    uint32_t lw4 = lds_base + 512 + lane * 4;
    uint32_t lx  = lds_base + WROW * 4 + lane * 16;
    asm volatile("global_load_async_to_lds_b128 %0, %1, off\n\t"
                 "global_load_async_to_lds_b32  %2, %3, off\n\t"
                 "global_load_async_to_lds_b128 %4, %5, off"
                 :
                 : "v"(lw), "v"(wg), "v"(lw4), "v"(w4g), "v"(lx), "v"(xg)
                 : "memory");
}

// ---------------- Main gather kernel: one wave32 per node ----------------

__global__ void __launch_bounds__(256)
node_gather_kernel(const float* __restrict__ x,
                   const float* __restrict__ ea,
                   const float* __restrict__ ew,
                   const int*   __restrict__ esrc,
                   const int*   __restrict__ offs,
                   const int*   __restrict__ edge_ids,
                   float* __restrict__ out, int N) {
    __shared__ __align__(16) float stage[WAVES_PER_BLOCK][2][STAGE_F];

    const int lane = threadIdx.x & 31;
    const int warp = threadIdx.x >> 5;
    const int node = blockIdx.x * (blockDim.x >> 5) + warp;
    if (node >= N) return;

    const int beg = offs[node];
    const int end = offs[node + 1];
    const int deg = end - beg;

    // Generic->LDS: low 32 bits of a flat shared-aperture address are the LDS
    // byte offset (ISA 10.2).
    const uint32_t lds0 = (uint32_t)(uintptr_t)(&stage[warp][0][0]);
    const uint32_t lds1 = (uint32_t)(uintptr_t)(&stage[warp][1][0]);

    __builtin_prefetch(edge_ids + beg + lane, 0, 0);   // warm the segment list

    float o0 = 0.f;
    float o1x = 0.f, o1y = 0.f, o1z = 0.f;
    float o2x = 0.f, o2y = 0.f, o2z = 0.f;
    float o3 = 0.f;
    float o4x = 0.f, o4y = 0.f, o4z = 0.f;

    // Deterministic order: iterate the segment in ascending edge-id order,
    // independent of the atomic slot order chosen by scatter_kernel.
    // (wave32 min-selection; __shfl_xor lowers to LDS-permute ops)
    auto select_min_gt = [&](int last) -> int {
        int m = 0x7FFFFFFF;
        for (int j = beg + lane; j < end; j += 32) {
            int id = edge_ids[j];
            if (id > last && id < m) m = id;
        }
        #pragma unroll
        for (int off = 16; off > 0; off >>= 1) {
            int o = __shfl_xor(m, off, 32);
            m = (o < m) ? o : m;
        }
        return m;
    };

    int e_cur = -1;
    if (deg > 0) {
        e_cur = select_min_gt(-1);
        const int s0 = esrc[e_cur];
        issue_stage(ew + (size_t)e_cur * WROW, x + (size_t)s0 * XROW, lds0, lane);
    }

    for (int it = 0; it < deg; ++it) {
        // Software pipeline: stage the NEXT edge while current transfers land.
        int e_next = -1;
        if (it + 1 < deg) {
            e_next = select_min_gt(e_cur);
            const int sn = esrc[e_next];
            issue_stage(ew + (size_t)e_next * WROW, x + (size_t)sn * XROW,
                        ((it + 1) & 1) ? lds1 : lds0, lane);
            // current edge's 3 transfers done, next edge's 3 may remain
            asm volatile("s_wait_asynccnt 0x3" ::: "memory");
        } else {
            asm volatile("s_wait_asynccnt 0x0" ::: "memory");
        }

        const float4 a = *reinterpret_cast<const float4*>(ea + (size_t)e_cur * 4);

        const float* wb = &stage[warp][it & 1][0];
        const float w0 = wb[lane];
        const float w1 = wb[32 + lane];
        const float w2 = wb[64 + lane];
        const float w3 = wb[96 + lane];
        const float w4 = wb[128 + lane];

        const float* xb = &stage[warp][it & 1][WROW];
        const float xs0 = xb[lane];
        const float xx = xb[MULC + 3 * lane + 0];
        const float xy = xb[MULC + 3 * lane + 1];
        const float xz = xb[MULC + 3 * lane + 2];

        // o0 = w0 * xs0 * a0
        o0 = fmaf(w0 * xs0, a.x, o0);
        // o1[k] = (w1 * xs0) * a1[k]
        const float c1 = w1 * xs0;
        o1x = fmaf(c1, a.y, o1x);
        o1y = fmaf(c1, a.z, o1y);
        o1z = fmaf(c1, a.w, o1z);
        // o2[k] = w2 * x1[k] * a0
        const float c2 = w2 * a.x;
        o2x = fmaf(c2, xx, o2x);
        o2y = fmaf(c2, xy, o2y);
        o2z = fmaf(c2, xz, o2z);
        // o3 = w3 * dot(x1, a1) / sqrt3   (scale applied after the loop)
        const float dot = fmaf(xx, a.y, fmaf(xy, a.z, xz * a.w));
        o3 = fmaf(w3, dot, o3);
        // o4[k] = w4 * cross(x1, a1)[k] / sqrt2
        const float cx = xy * a.w - xz * a.z;
        const float cy = xz * a.y - xx * a.w;
        const float cz = xx * a.z - xy * a.y;
        o4x = fmaf(w4, cx, o4x);
        o4y = fmaf(w4, cy, o4y);
        o4z = fmaf(w4, cz, o4z);

        e_cur = e_next;
    }

    float* orow = out + (size_t)node * NFEAT;
    orow[lane]               = o0;
    orow[32  + 3 * lane + 0] = o1x;
    orow[32  + 3 * lane + 1] = o1y;
    orow[32  + 3 * lane + 2] = o1z;
    orow[128 + 3 * lane + 0] = o2x;
    orow[128 + 3 * lane + 1] = o2y;
    orow[128 + 3 * lane + 2] = o2z;
    orow[224 + lane]         = o3 * inv_sqrt3();
    orow[256 + 3 * lane + 0] = o4x * inv_sqrt2();
    orow[256 + 3 * lane + 1] = o4y * inv_sqrt2();
    orow[256 + 3 * lane + 2] = o4z * inv_sqrt2();
}

// ---------------- Fallback path (workspace too small): float atomics ----------------

__global__ void zero_kernel(float* __restrict__ out, int n) {
    int i = blockIdx.x * blockDim.x + threadIdx.x;
    int stride = gridDim.x * blockDim.x;
    for (; i < n; i += stride) out[i] = 0.f;
}

__global__ void __launch_bounds__(256)
edge_atomic_kernel(const float* __restrict__ x,
                   const float* __restrict__ ea,
                   const float* __restrict__ ew,
                   const int*   __restrict__ edst,
                   const int*   __restrict__ esrc,
                   float* __restrict__ out, int E) {
    const int lane = threadIdx.x & 31;
    const int e = blockIdx.x * (blockDim.x >> 5) + (threadIdx.x >> 5);
    if (e >= E) return;

    const int s = esrc[e];
    const int d = edst[e];
    const float4 a = *reinterpret_cast<const float4*>(ea + (size_t)e * 4);
    const float* wr = ew + (size_t)e * WROW + lane;
    const float w0 = wr[0], w1 = wr[32], w2 = wr[64], w3 = wr[96], w4 = wr[128];
    const float* xr = x + (size_t)s * XROW;
    const float xs0 = xr[lane];
    const float* x1 = xr + MULC + 3 * lane;
    const float xx = x1[0], xy = x1[1], xz = x1[2];

    const float c1 = w1 * xs0;
    const float c2 = w2 * a.x;
    const float dot = fmaf(xx, a.y, fmaf(xy, a.z, xz * a.w));
    const float cx = xy * a.w - xz * a.z;
    const float cy = xz * a.y - xx * a.w;
    const float cz = xx * a.z - xy * a.y;

    float* orow = out + (size_t)d * NFEAT;
    atomicAdd(&orow[lane],               w0 * xs0 * a.x);
    atomicAdd(&orow[32  + 3 * lane + 0], c1 * a.y);
    atomicAdd(&orow[32  + 3 * lane + 1], c1 * a.z);
    atomicAdd(&orow[32  + 3 * lane + 2], c1 * a.w);
    atomicAdd(&orow[128 + 3 * lane + 0], c2 * xx);
    atomicAdd(&orow[128 + 3 * lane + 1], c2 * xy);
    atomicAdd(&orow[128 + 3 * lane + 2], c2 * xz);
    atomicAdd(&orow[224 + lane],         w3 * dot * inv_sqrt3());
    atomicAdd(&orow[256 + 3 * lane + 0], w4 * cx * inv_sqrt2());
    atomicAdd(&orow[256 + 3 * lane + 1], w4 * cy * inv_sqrt2());
    atomicAdd(&orow[256 + 3 * lane + 2], w4 * cz * inv_sqrt2());
}

// ---------------- launch ----------------

extern "C" void kernel_launch(void* const* d_in, const int* in_sizes, int n_in,
                              void* d_out, int out_size, void* d_ws, size_t ws_size,
                              hipStream_t stream) {
    const float* x    = (const float*)d_in[0];
    const float* ea   = (const float*)d_in[1];
    const float* ew   = (const float*)d_in[2];
    const int*   edst = (const int*)d_in[3];
    const int*   esrc = (const int*)d_in[4];
    float* out = (float*)d_out;

    const int N = in_sizes[0] / XROW;
    const int E = in_sizes[3];

    const size_t need = (size_t)(2 * (size_t)N + 1 + (size_t)E) * sizeof(int);
    if (ws_size >= need) {
        int* hist_offs = (int*)d_ws;              // N+1
        int* cursor    = hist_offs + (N + 1);     // N
        int* edge_ids  = cursor + N;              // E

        hipMemsetAsync(hist_offs, 0, (size_t)N * sizeof(int), stream);
        hist_kernel<<<(E + 255) / 256, 256, 0, stream>>>(edst, hist_offs, E);
        scan_kernel<<<1, 1024, 0, stream>>>(hist_offs, cursor, N);
        scatter_kernel<<<(E + 255) / 256, 256, 0, stream>>>(edst, cursor, edge_ids, E);
        node_gather_kernel<<<(N + WAVES_PER_BLOCK - 1) / WAVES_PER_BLOCK, 256, 0, stream>>>(
            x, ea, ew, esrc, hist_offs, edge_ids, out, N);
    } else {
        zero_kernel<<<4096, 256, 0, stream>>>(out, out_size);
        edge_atomic_kernel<<<(E + 7) / 8, 256, 0, stream>>>(x, ea, ew, edst, esrc, out, E);
    }
}